// PointsRenderLayer_72885595013752
// MI455X (gfx1250) — compile-verified
//
#include <hip/hip_runtime.h>

typedef __attribute__((ext_vector_type(2))) float v2f;
typedef __attribute__((ext_vector_type(8))) float v8f;
typedef __attribute__((ext_vector_type(4))) unsigned int u32x4;
typedef __attribute__((ext_vector_type(8))) int i32x8;
typedef __attribute__((ext_vector_type(4))) int i32x4;

#define RADIUS   0.05f
#define R2       (RADIUS * RADIUS)
#define INV_R2   (1.0f / (RADIUS * RADIUS))
#define KP       8      // points_per_pixel
#define CF       8      // feature channels
#define IMG      128    // H == W
#define NMAX     4096   // point capacity staged in LDS
#define ZSENT    3.0e30f
#define NWAVE    4      // waves per block

// ---------------------------------------------------------------------------
// Kernel 1: project points once into L2-resident scratch as
//   { -2*px, -2*py, px*px + py*py, z }  with px = x/z, py = y/z
// ---------------------------------------------------------------------------
__global__ void points_preprocess_kernel(const float* __restrict__ pts,
                                         float4* __restrict__ ws, int N) {
    int i = blockIdx.x * blockDim.x + threadIdx.x;
    if (i >= N) return;
    float x = pts[3 * i + 0];
    float y = pts[3 * i + 1];
    float z = pts[3 * i + 2];
    float invz = 1.0f / z;
    float px = x * invz;
    float py = y * invz;
    float4 o;
    o.x = -2.0f * px;
    o.y = -2.0f * py;
    o.z = px * px + py * py;
    o.w = z;
    ws[i] = o;
}

// ---------------------------------------------------------------------------
// Kernel 2: rasterize. One wave = 32 pixels (two 16-pixel groups). Per
// 16-point tile, TWO independent V_WMMA_F32_16X16X4_F32 share one B operand:
//   D[m][n] = gx_m*(-2px_n) + gy_m*(-2py_n) + 1*pp_n + gg_m*1 = dist2
// Point table (64 KB) staged into LDS once per block by the Tensor Data
// Mover; the next tile's ds_load_b128 is issued before the current WMMAs so
// LDS latency overlaps matrix work. Every lane owns one pixel (lane = pixel
// base+lane), so slow-path top-K insertion and compositing use full EXEC.
// ---------------------------------------------------------------------------
__global__ __launch_bounds__(128) void points_render_kernel(
    const float4* __restrict__ P4,     // preprocessed points [N] in d_ws
    const float*  __restrict__ feat,   // [N, CF]
    float*        __restrict__ out,    // [IMG*IMG, CF]
    int N) {
    __shared__ float4 ptsLDS[NMAX];            // 64 KB staged point table
    __shared__ float  tileD[NWAVE][2][16][16]; // 8 KB: [wave][grp][n][m]

    const int tid  = threadIdx.x;
    const int wave = tid >> 5;
    const int lane = tid & 31;
    const int l16  = lane & 15;
    const int half = lane >> 4;

    // ---- TDM stage: one DMA moves the whole point table into LDS ----------
    if (wave == 0) {
        // flat-shared address low 32 bits == LDS byte offset (ISA 10.2)
        unsigned int lds_base = (unsigned int)(uintptr_t)&ptsLDS[0];
        unsigned long long ga = (unsigned long long)(uintptr_t)P4;
        unsigned int nd0 = (unsigned int)N * 4u;  // elements (4B units) in dim0

        u32x4 g0;
        g0[0] = 1u;                                   // count=1, user mode
        g0[1] = lds_base;                             // lds_addr [63:32]
        g0[2] = (unsigned int)(ga & 0xffffffffull);   // global_addr [95:64]
        g0[3] = (unsigned int)((ga >> 32) & 0x01ffffffull) | (2u << 30); // type=2

        i32x8 g1;
        g1[0] = (int)(2u << 16);                      // data_size=4B, wg_mask=0
        g1[1] = (int)((nd0 & 0xffffu) << 16);         // tensor_dim0[15:0]
        g1[2] = (int)(((nd0 >> 16) & 0xffffu) | (1u << 16)); // dim0 hi, dim1=1
        g1[3] = (int)((nd0 & 0xffffu) << 16);         // tile_dim0 = nd0
        g1[4] = 1;                                    // tile_dim1=1, tile_dim2=0
        g1[5] = (int)nd0;                             // tensor_dim0_stride
        g1[6] = 0;
        g1[7] = 0;
        i32x4 gz4 = {0, 0, 0, 0};                     // groups 2/3 unused (<=2D)
        i32x8 gz8 = {0, 0, 0, 0, 0, 0, 0, 0};

        __builtin_amdgcn_tensor_load_to_lds(g0, g1, gz4, gz4, gz8, 0);
        __builtin_amdgcn_s_wait_tensorcnt(0);
    }
    __syncthreads();

    const int base = (blockIdx.x * NWAVE + wave) * 32;  // 32 pixels per wave
    // pixel coords for A0 (pixels base+l16) and A1 (pixels base+16+l16)
    const int pix0 = base + l16;
    const int pix1 = base + 16 + l16;
    const float gx0 = ((float)(pix0 & (IMG - 1)) + 0.5f) * (2.0f / IMG) - 1.0f;
    const float gy0 = ((float)(pix0 >> 7)        + 0.5f) * (2.0f / IMG) - 1.0f;
    const float gx1 = ((float)(pix1 & (IMG - 1)) + 0.5f) * (2.0f / IMG) - 1.0f;
    const float gy1 = ((float)(pix1 >> 7)        + 0.5f) * (2.0f / IMG) - 1.0f;
    const float gg0 = gx0 * gx0 + gy0 * gy0;
    const float gg1 = gx1 * gx1 + gy1 * gy1;

    // A matrices: K = {gx, gy, 1, gg}; lanes 0-15 carry K0/K1, 16-31 K2/K3
    v2f A0, A1;
    A0.x = half ? 1.0f : gx0;  A0.y = half ? gg0 : gy0;
    A1.x = half ? 1.0f : gx1;  A1.y = half ? gg1 : gy1;

    // per-pixel K-list: lane owns pixel base+lane (group = half)
    float zk[KP], d2k[KP];
    int   idk[KP];
#pragma unroll
    for (int k = 0; k < KP; ++k) { zk[k] = ZSENT; d2k[k] = 0.0f; idk[k] = 0; }

    const int T = N >> 4;  // 16-point tiles
    float4 Pt = ptsLDS[l16];
    for (int t = 0; t < T; ++t) {
        // issue next tile's load early so DS latency overlaps the WMMAs
        const int tn = (t + 1 < T) ? (t + 1) : t;
        float4 Ptn = ptsLDS[(tn << 4) + l16];

        v2f Bmat;
        Bmat.x = half ? Pt.z : Pt.x;          // K=2 row (pp) : K=0 row (-2px)
        Bmat.y = half ? 1.0f : Pt.y;          // K=3 row (1)  : K=1 row (-2py)

        v8f c = {};
        v8f d0 = __builtin_amdgcn_wmma_f32_16x16x4_f32(
            false, A0, false, Bmat, (short)0, c, false, false);
        v8f d1 = __builtin_amdgcn_wmma_f32_16x16x4_f32(
            false, A1, false, Bmat, (short)0, c, false, false);

        float mn0 = fminf(fminf(fminf(d0[0], d0[1]), fminf(d0[2], d0[3])),
                          fminf(fminf(d0[4], d0[5]), fminf(d0[6], d0[7])));
        float mn1 = fminf(fminf(fminf(d1[0], d1[1]), fminf(d1[2], d1[3])),
                          fminf(fminf(d1[4], d1[5]), fminf(d1[6], d1[7])));
        if (__builtin_amdgcn_ballot_w32(fminf(mn0, mn1) < R2) != 0u) {
            // rare path (~0.4%/tile-pair): spill both tiles, insert per pixel
#pragma unroll
            for (int r = 0; r < 8; ++r) {
                tileD[wave][0][l16][half * 8 + r] = d0[r];  // [n][m] contiguous
                tileD[wave][1][l16][half * 8 + r] = d1[r];
            }
            asm volatile("s_wait_dscnt 0x0" ::: "memory");
            // every lane scans its own pixel's row (group = half, m = l16)
#pragma unroll 1
            for (int n = 0; n < 16; ++n) {
                float d2 = tileD[wave][half][n][l16];
                float pz = ptsLDS[(t << 4) + n].w;
                if (d2 < R2 && pz > 0.0f && pz < zk[KP - 1]) {
                    float cz = pz, cd = d2;
                    int   ci = (t << 4) + n;
#pragma unroll
                    for (int k = 0; k < KP; ++k) {
                        if (cz < zk[k]) {
                            float tz = zk[k];  zk[k]  = cz; cz = tz;
                            float td = d2k[k]; d2k[k] = cd; cd = td;
                            int   ti = idk[k]; idk[k] = ci; ci = ti;
                        }
                    }
                }
            }
            asm volatile("s_wait_dscnt 0x0" ::: "memory");
        }
        Pt = Ptn;
    }

    // front-to-back alpha compositing, all 32 lanes (pixel = base+lane)
    {
        const int pix = base + lane;
        float acc[CF];
#pragma unroll
        for (int c0 = 0; c0 < CF; ++c0) acc[c0] = 0.0f;
        float trans = 1.0f;
#pragma unroll
        for (int k = 0; k < KP; ++k) {
            bool  v     = zk[k] < 1.0e30f;
            float alpha = v ? (1.0f - d2k[k] * INV_R2) : 0.0f;
            alpha = fminf(fmaxf(alpha, 0.0f), 1.0f);
            float w = alpha * trans;
            trans *= (1.0f - alpha);
            const float4* f = (const float4*)(feat + (size_t)idk[k] * CF);
            float4 f0 = f[0];
            float4 f1 = f[1];
            acc[0] += w * f0.x; acc[1] += w * f0.y;
            acc[2] += w * f0.z; acc[3] += w * f0.w;
            acc[4] += w * f1.x; acc[5] += w * f1.y;
            acc[6] += w * f1.z; acc[7] += w * f1.w;
        }
        float4* o = (float4*)(out + (size_t)pix * CF);
        o[0] = make_float4(acc[0], acc[1], acc[2], acc[3]);
        o[1] = make_float4(acc[4], acc[5], acc[6], acc[7]);
    }
}

// ---------------------------------------------------------------------------
extern "C" void kernel_launch(void* const* d_in, const int* in_sizes, int n_in,
                              void* d_out, int out_size, void* d_ws, size_t ws_size,
                              hipStream_t stream) {
    const float* pts  = (const float*)d_in[0];  // [B,N,3] f32
    const float* feat = (const float*)d_in[1];  // [B,N,C] f32
    const int N = in_sizes[0] / 3;              // 4096

    float4* ws4 = (float4*)d_ws;                // 64 KB scratch, L2-resident
    points_preprocess_kernel<<<(N + 255) / 256, 256, 0, stream>>>(pts, ws4, N);

    const int P      = IMG * IMG;        // 16384 pixels
    const int blocks = P / (NWAVE * 32); // 128 blocks, 32 pixels per wave
    points_render_kernel<<<blocks, NWAVE * 32, 0, stream>>>(ws4, feat,
                                                            (float*)d_out, N);
}